// GCN_62208306315756
// MI455X (gfx1250) — compile-verified
//
#include <hip/hip_runtime.h>
#include <cstdint>
#include <cstddef>

typedef __attribute__((ext_vector_type(2))) float v2f;
typedef __attribute__((ext_vector_type(8))) float v8f;

#define HID 100
#define LDH 112      // padded feature-row stride (7 tiles of 16)
#define NGRAPH 64
#define MT 4         // M-tiles (of 16 rows) per wave in the WMMA GEMM

// ---------------- degree / symmetric norm ----------------
__global__ void k_deg_init(float* deg, int n) {
  int i = blockIdx.x * blockDim.x + threadIdx.x;
  if (i < n) deg[i] = 1.0f;                    // self loop contributes 1
}
__global__ void k_deg_accum(const int* __restrict__ dst, int nE, float* deg) {
  int e = blockIdx.x * blockDim.x + threadIdx.x;
  if (e < nE) atomicAdd(&deg[dst[e]], 1.0f);
}
__global__ void k_deg_to_dinv(float* deg, int n) {
  int i = blockIdx.x * blockDim.x + threadIdx.x;
  if (i < n) { float d = deg[i]; deg[i] = (d > 0.f) ? rsqrtf(d) : 0.f; }
}

// ---------------- f32 WMMA GEMM: C[M x nReal] = A[M x K] * B[K x nReal] ----
// One wave per (MT*16) x 16 output slab: MT independent accumulators share one
// B fragment per K-step (register-level B reuse + independent WMMA chains).
__global__ void k_wmma_gemm(const float* __restrict__ A, int lda, int K,
                            const float* __restrict__ B, int ldb, int nReal,
                            float* __restrict__ C, int ldc, int M) {
  const int lane = threadIdx.x;          // 0..31, wave32
  const bool hi  = lane >= 16;
  const int  l   = lane & 15;
  const int  mBase0 = blockIdx.x * (MT * 16);
  const int  n = blockIdx.y * 16 + l;    // B/C column for this lane
  const bool nOK = (n < nReal);

  // 16x4 f32 A layout: lanes 0-15 hold K=0,1 in v0,v1; lanes 16-31 hold K=2,3.
  const float* Arow[MT];
#pragma unroll
  for (int t = 0; t < MT; ++t) {
    int m  = mBase0 + t * 16 + l;
    int mc = (m < M) ? m : (M - 1);      // clamp reads; stores are guarded
    Arow[t] = A + (size_t)mc * lda + (hi ? 2 : 0);
  }
  // 4x16 f32 B layout (transpose-symmetric): lanes 0-15 rows K=0,1; 16-31 rows K=2,3.
  const float* Bcol = B + (nOK ? n : 0);
  const size_t kofs = (size_t)(hi ? 2 : 0) * (size_t)ldb;

  v8f acc[MT];
#pragma unroll
  for (int t = 0; t < MT; ++t) acc[t] = v8f{};

  for (int k = 0; k < K; k += 4) {
    size_t kb = (size_t)k * (size_t)ldb + kofs;
    v2f b;
    float b0 = Bcol[kb];
    float b1 = Bcol[kb + ldb];
    b.x = nOK ? b0 : 0.0f;
    b.y = nOK ? b1 : 0.0f;
    v2f a[MT];
#pragma unroll
    for (int t = 0; t < MT; ++t) {       // issue all A loads first
      a[t].x = Arow[t][k];
      a[t].y = Arow[t][k + 1];
    }
#pragma unroll
    for (int t = 0; t < MT; ++t) {       // 4 independent WMMAs back-to-back
      acc[t] = __builtin_amdgcn_wmma_f32_16x16x4_f32(
          false, a[t], false, b, (short)0, acc[t], false, false);
    }
  }

  // C/D layout: VGPR r -> row (tile*16 + r) lanes 0-15, (tile*16 + 8 + r) lanes 16-31.
  if (nOK) {
#pragma unroll
    for (int t = 0; t < MT; ++t) {
      int rowBase = mBase0 + t * 16 + (hi ? 8 : 0);
#pragma unroll
      for (int r = 0; r < 8; ++r) {
        int row = rowBase + r;
        if (row < M) C[(size_t)row * ldc + n] = acc[t][r];
      }
    }
  }
}

// ---------------- aggregation ----------------
// agg[n][f] = h[n][f] * dinv[n]^2   (self-loop term, also initializes buffer)
__global__ void k_agg_self(const float* __restrict__ h, const float* __restrict__ dinv,
                           float* __restrict__ agg) {
  int n = blockIdx.x;
  int f = threadIdx.x;
  if (f >= HID) return;
  float di = dinv[n];
  agg[(size_t)n * LDH + f] = h[(size_t)n * LDH + f] * di * di;
}

// agg[d][f] += h[s][f] * dinv[s]*dinv[d]  over all edges
__global__ void k_agg_edges(const int* __restrict__ src, const int* __restrict__ dst,
                            const float* __restrict__ h, const float* __restrict__ dinv,
                            float* __restrict__ agg) {
  int e = blockIdx.x;
  int f = threadIdx.x;
  if (f >= HID) return;
  int s = src[e];
  int d = dst[e];
  float norm = dinv[s] * dinv[d];
  atomicAdd(&agg[(size_t)d * LDH + f], h[(size_t)s * LDH + f] * norm);
}

__global__ void k_bias_tanh(float* __restrict__ a, const float* __restrict__ b) {
  int n = blockIdx.x;
  int f = threadIdx.x;
  if (f >= HID) return;
  size_t idx = (size_t)n * LDH + f;
  a[idx] = tanhf(a[idx] + b[f]);
}
__global__ void k_bias_add(float* __restrict__ a, const float* __restrict__ b) {
  int n = blockIdx.x;
  int f = threadIdx.x;
  if (f >= HID) return;
  size_t idx = (size_t)n * LDH + f;
  a[idx] = a[idx] + b[f];
}

// ---------------- pooling + head ----------------
__global__ void k_pool_zero(float* __restrict__ sums, float* __restrict__ cnts) {
  int i = blockIdx.x * blockDim.x + threadIdx.x;
  if (i < NGRAPH * HID) sums[i] = 0.0f;
  if (i < NGRAPH) cnts[i] = 0.0f;
}
__global__ void k_pool_accum(const float* __restrict__ a, const int* __restrict__ batch,
                             float* __restrict__ sums, float* __restrict__ cnts) {
  int n = blockIdx.x;
  int f = threadIdx.x;
  if (f >= HID) return;
  int g = batch[n];
  atomicAdd(&sums[g * HID + f], a[(size_t)n * LDH + f]);
  if (f == 0) atomicAdd(&cnts[g], 1.0f);
}
__global__ void k_head(const float* __restrict__ sums, const float* __restrict__ cnts,
                       const float* __restrict__ Wlin, const float* __restrict__ blin,
                       float* __restrict__ out) {
  int g = threadIdx.x;
  if (g >= NGRAPH) return;
  float c = fmaxf(cnts[g], 1.0f);
  float acc = 0.0f;
  for (int f = 0; f < HID; ++f) acc += (sums[g * HID + f] / c) * Wlin[f];
  out[g] = acc + blin[0];
}

extern "C" void kernel_launch(void* const* d_in, const int* in_sizes, int n_in,
                              void* d_out, int out_size, void* d_ws, size_t ws_size,
                              hipStream_t stream) {
  const float* x     = (const float*)d_in[0];     // [N, 128]
  const int*   eidx  = (const int*)d_in[1];       // [2, E] flat
  const int*   batch = (const int*)d_in[2];       // [N]
  const float* W1    = (const float*)d_in[3];     // [128, 100]
  const float* b1    = (const float*)d_in[4];     // [100]
  const float* W2    = (const float*)d_in[5];     // [100, 100]
  const float* b2    = (const float*)d_in[6];     // [100]
  const float* Wlin  = (const float*)d_in[7];     // [100, 1]
  const float* blin  = (const float*)d_in[8];     // [1]
  float* out = (float*)d_out;                     // [64]

  const int N   = in_sizes[2];                    // 50000
  const int E   = in_sizes[1] / 2;                // 800000
  const int DIN = in_sizes[3] / HID;              // 128
  const int* src = eidx;
  const int* dst = eidx + E;

  // workspace carve-up (floats)
  float* ws   = (float*)d_ws;
  size_t off  = 0;
  float* dinv = ws + off; off += ((size_t)N + 127) & ~(size_t)127;
  float* bufA = ws + off; off += (size_t)N * LDH;   // activations / aggregated
  float* bufB = ws + off; off += (size_t)N * LDH;   // dense-transform output
  float* sums = ws + off; off += (size_t)NGRAPH * HID;
  float* cnts = ws + off; off += NGRAPH;
  (void)ws_size; (void)n_in; (void)out_size;

  const int mTiles   = (N + 15) / 16;                 // 3125
  const int mGroups  = (mTiles + MT - 1) / MT;        // 782
  const int nTiles   = (HID + 15) / 16;               // 7
  dim3 gemmGrid(mGroups, nTiles);

  // ---- degrees -> dinv ----
  k_deg_init   <<<(N + 255) / 256, 256, 0, stream>>>(dinv, N);
  k_deg_accum  <<<(E + 255) / 256, 256, 0, stream>>>(dst, E, dinv);
  k_deg_to_dinv<<<(N + 255) / 256, 256, 0, stream>>>(dinv, N);

  // ---- layer 1: h1 = x @ W1 ; aggregate ; +b1 ; tanh ----
  k_wmma_gemm<<<gemmGrid, 32, 0, stream>>>(x, DIN, DIN, W1, HID, HID, bufB, LDH, N);
  k_agg_self <<<N, 128, 0, stream>>>(bufB, dinv, bufA);
  k_agg_edges<<<E, 128, 0, stream>>>(src, dst, bufB, dinv, bufA);
  k_bias_tanh<<<N, 128, 0, stream>>>(bufA, b1);

  // ---- layer 2: h2 = a1 @ W2 ; aggregate ; +b2 ----
  k_wmma_gemm<<<gemmGrid, 32, 0, stream>>>(bufA, LDH, HID, W2, HID, HID, bufB, LDH, N);
  k_agg_self <<<N, 128, 0, stream>>>(bufB, dinv, bufA);
  k_agg_edges<<<E, 128, 0, stream>>>(src, dst, bufB, dinv, bufA);
  k_bias_add <<<N, 128, 0, stream>>>(bufA, b2);

  // ---- mean pool + linear head ----
  k_pool_zero <<<(NGRAPH * HID + 255) / 256, 256, 0, stream>>>(sums, cnts);
  k_pool_accum<<<N, 128, 0, stream>>>(bufA, batch, sums, cnts);
  k_head      <<<1, 64, 0, stream>>>(sums, cnts, Wlin, blin, out);
}